// RWKV7Attention_42228118454503
// MI455X (gfx1250) — compile-verified
//
#include <hip/hip_runtime.h>
#include <hip/hip_bf16.h>
#include <math.h>

// Problem constants (fixed by the reference)
#define BB   2
#define TTT  2048
#define CCC  1024
#define HHH  16
#define DDD  64
#define MMM  (BB * TTT)      // 4096 rows
#define EPSV 1e-5f

typedef __attribute__((ext_vector_type(2))) float v2f;
typedef __attribute__((ext_vector_type(8))) float v8f;

template <int ACT>
__device__ __forceinline__ float act_apply(float vv)
{
    if (ACT == 1) return tanhf(vv);
    if (ACT == 2) return 1.f / (1.f + expf(-vv));
    if (ACT == 3) return 0.60653065971263342f / (1.f + expf(-vv)); // sigmoid * e^-0.5
    return vv;
}

// ---------------------------------------------------------------------------
// Fused WMMA fp32 GEMM:  Y[M,N] = act( X'[M,K] @ W[K,N] + bias )
//   MIX=1 : X' = hidden + (shift(hidden) - hidden) * mix   (RWKV time-shift)
//           implemented branchlessly: the t==0 row loads itself as the
//           "previous" row and scales its contribution by sel=0.
//   MIX=0 : X' = X
// One wave computes a 32x64 strip: 2 A fragments x 4 B fragments feed
// 8 V_WMMA_F32_16X16X4_F32 per k-step (A reused 4x, B reused 2x).
// Block = 256 threads = 8 waves stacked along M (block tile 256 x 64).
// ---------------------------------------------------------------------------
template <int MIX, int ACT>
__global__ __launch_bounds__(256) void gemm16_wmma(
    const float* __restrict__ X, const float* __restrict__ mix,
    const float* __restrict__ W, const float* __restrict__ bias,
    float* __restrict__ Y, int M, int K, int N)
{
    const int wave = threadIdx.x >> 5;
    const int lane = threadIdx.x & 31;
    const int half = lane >> 4;      // 0: K pair {0,1}, 1: K pair {2,3}
    const int l16  = lane & 15;
    const int tileM  = (blockIdx.y * 8 + wave) * 32;
    const int tileN0 = blockIdx.x * 64;
    const int mrow0  = tileM + l16;       // A rows this lane loads
    const int mrow1  = tileM + 16 + l16;

    // branchless time-shift selectors (per A row, computed once)
    const ptrdiff_t poff0 = (MIX && (mrow0 % TTT) != 0) ? (ptrdiff_t)K : 0;
    const ptrdiff_t poff1 = (MIX && (mrow1 % TTT) != 0) ? (ptrdiff_t)K : 0;
    const float     sel0  = (MIX && (mrow0 % TTT) != 0) ? 1.f : 0.f;
    const float     sel1  = (MIX && (mrow1 % TTT) != 0) ? 1.f : 0.f;

    const v8f vzero = {0.f, 0.f, 0.f, 0.f, 0.f, 0.f, 0.f, 0.f};
    v8f a0 = vzero, a1 = vzero, a2 = vzero, a3 = vzero;   // rows tileM..tileM+15
    v8f c0 = vzero, c1 = vzero, c2 = vzero, c3 = vzero;   // rows tileM+16..tileM+31

    const float* Wp = W + tileN0 + l16;   // this lane's column within the strip
    const float* X0 = X + (size_t)mrow0 * K;
    const float* X1 = X + (size_t)mrow1 * K;
    const float* P0 = X0 - poff0;         // previous row (or self when t==0)
    const float* P1 = X1 - poff1;

    for (int k0 = 0; k0 < K; k0 += 4) {
        const int k = k0 + 2 * half;
        v2f av0, av1;
        if (MIX) {
            const float m0 = mix[k], m1 = mix[k + 1];
            const float h00 = X0[k], h01 = X0[k + 1];
            const float h10 = X1[k], h11 = X1[k + 1];
            av0.x = h00 + (sel0 * P0[k]     - h00) * m0;
            av0.y = h01 + (sel0 * P0[k + 1] - h01) * m1;
            av1.x = h10 + (sel1 * P1[k]     - h10) * m0;
            av1.y = h11 + (sel1 * P1[k + 1] - h11) * m1;
        } else {
            av0.x = X0[k]; av0.y = X0[k + 1];
            av1.x = X1[k]; av1.y = X1[k + 1];
        }
        const float* w0 = Wp + (size_t)k * N;        // row k
        const float* w1 = Wp + (size_t)(k + 1) * N;  // row k+1
        v2f b0; b0.x = w0[0];  b0.y = w1[0];
        v2f b1; b1.x = w0[16]; b1.y = w1[16];
        v2f b2; b2.x = w0[32]; b2.y = w1[32];
        v2f b3; b3.x = w0[48]; b3.y = w1[48];
        a0 = __builtin_amdgcn_wmma_f32_16x16x4_f32(false, av0, false, b0, (short)0, a0, false, false);
        a1 = __builtin_amdgcn_wmma_f32_16x16x4_f32(false, av0, false, b1, (short)0, a1, false, false);
        a2 = __builtin_amdgcn_wmma_f32_16x16x4_f32(false, av0, false, b2, (short)0, a2, false, false);
        a3 = __builtin_amdgcn_wmma_f32_16x16x4_f32(false, av0, false, b3, (short)0, a3, false, false);
        c0 = __builtin_amdgcn_wmma_f32_16x16x4_f32(false, av1, false, b0, (short)0, c0, false, false);
        c1 = __builtin_amdgcn_wmma_f32_16x16x4_f32(false, av1, false, b1, (short)0, c1, false, false);
        c2 = __builtin_amdgcn_wmma_f32_16x16x4_f32(false, av1, false, b2, (short)0, c2, false, false);
        c3 = __builtin_amdgcn_wmma_f32_16x16x4_f32(false, av1, false, b3, (short)0, c3, false, false);
    }

    float bi0 = 0.f, bi1 = 0.f, bi2 = 0.f, bi3 = 0.f;
    if (bias) {
        bi0 = bias[tileN0 + l16];
        bi1 = bias[tileN0 + 16 + l16];
        bi2 = bias[tileN0 + 32 + l16];
        bi3 = bias[tileN0 + 48 + l16];
    }
    float* Yp = Y + tileN0 + l16;
#pragma unroll
    for (int i = 0; i < 8; ++i) {
        const int r0 = tileM + half * 8 + i;   // C/D layout: rows i / i+8
        const size_t rb0 = (size_t)r0 * N;
        const size_t rb1 = (size_t)(r0 + 16) * N;
        Yp[rb0]      = act_apply<ACT>(a0[i] + bi0);
        Yp[rb0 + 16] = act_apply<ACT>(a1[i] + bi1);
        Yp[rb0 + 32] = act_apply<ACT>(a2[i] + bi2);
        Yp[rb0 + 48] = act_apply<ACT>(a3[i] + bi3);
        Yp[rb1]      = act_apply<ACT>(c0[i] + bi0);
        Yp[rb1 + 16] = act_apply<ACT>(c1[i] + bi1);
        Yp[rb1 + 32] = act_apply<ACT>(c2[i] + bi2);
        Yp[rb1 + 48] = act_apply<ACT>(c3[i] + bi3);
    }
}

// ---------------------------------------------------------------------------
// Prep: per head-row (b,t,h) of 64 channels:
//   kk  = normalize(k * k_k)  (L2 over D, clamped at 1e-12)
//   bb  = kk * a
//   k  := k * (1 + (a-1)*k_a)   (in place -> becomes "k2")
// ---------------------------------------------------------------------------
__global__ __launch_bounds__(64) void rwkv7_prep_kernel(
    float* __restrict__ kbuf, const float* __restrict__ a,
    const float* __restrict__ k_k, const float* __restrict__ k_a,
    float* __restrict__ kk_out, float* __restrict__ bb_out)
{
    const int hr = blockIdx.x;          // 0 .. M*H-1
    const int h  = hr % HHH;
    const int m  = hr / HHH;
    const int d  = threadIdx.x;
    const int c  = h * DDD + d;
    const size_t idx = (size_t)m * CCC + c;

    __shared__ float red[DDD];
    const float kv  = kbuf[idx];
    const float av  = a[idx];
    const float kkv = kv * k_k[c];
    red[d] = kkv * kkv;
    __syncthreads();
    for (int s = 32; s > 0; s >>= 1) {
        if (d < s) red[d] += red[d + s];
        __syncthreads();
    }
    const float nrm = fmaxf(sqrtf(red[0]), 1e-12f);
    const float kkn = kkv / nrm;
    kk_out[idx] = kkn;
    bb_out[idx] = kkn * av;
    kbuf[idx]   = kv * (1.f + (av - 1.f) * k_a[c]);
}

// ---------------------------------------------------------------------------
// gfx1250 async global -> LDS copy (ASYNCcnt-tracked; see CDNA5 ISA)
// ---------------------------------------------------------------------------
__device__ __forceinline__ void async_copy_b32(float* lds_ptr, const float* gptr)
{
    const unsigned lds_off = (unsigned)(unsigned long long)lds_ptr; // low 32b = LDS addr
    asm volatile("global_load_async_to_lds_b32 %0, %1, off"
                 :: "v"(lds_off), "v"(gptr)
                 : "memory");
}

// ---------------------------------------------------------------------------
// Sequential RWKV7 scan. One block per (b,h); 64 lanes; lane i keeps row i of
// the 64x64 state S in 64 VGPRs. Per-step vectors are double-buffered in LDS
// via async global->LDS copies: while step t computes, step t+1's six
// D-vectors stream in. ASYNCcnt completes in issue order, so waiting for
// count<=6 (the 6 newest = step t+1) guarantees step t's data has landed.
//   sa  = -(S @ kk)                       (a_t = -kk)
//   S   = S*diag(ew) + sa b^T + v k^T     (ew = exp(log_w) precomputed)
//   o   = S @ r
// ---------------------------------------------------------------------------
__global__ __launch_bounds__(64) void rwkv7_scan_kernel(
    const float* __restrict__ r, const float* __restrict__ ew,
    const float* __restrict__ k, const float* __restrict__ v,
    const float* __restrict__ kk, const float* __restrict__ bb,
    float* __restrict__ o)
{
    const int bh = blockIdx.x;
    const int b  = bh / HHH;
    const int h  = bh % HHH;
    const int i  = threadIdx.x;

    float S[DDD];
#pragma unroll
    for (int j = 0; j < DDD; ++j) S[j] = 0.f;

    __shared__ float buf[2][6][DDD];

    // element index this lane copies / owns
    const size_t hbase = (size_t)b * TTT * CCC + (size_t)h * DDD + i;

    // kick off step 0
    async_copy_b32(&buf[0][0][i], r  + hbase);
    async_copy_b32(&buf[0][1][i], ew + hbase);
    async_copy_b32(&buf[0][2][i], k  + hbase);
    async_copy_b32(&buf[0][3][i], v  + hbase);
    async_copy_b32(&buf[0][4][i], kk + hbase);
    async_copy_b32(&buf[0][5][i], bb + hbase);

    size_t off = hbase;
    for (int t = 0; t < TTT; ++t, off += CCC) {
        const int par = t & 1;
        if (t + 1 < TTT) {
            const size_t noff = off + CCC;
            async_copy_b32(&buf[par ^ 1][0][i], r  + noff);
            async_copy_b32(&buf[par ^ 1][1][i], ew + noff);
            async_copy_b32(&buf[par ^ 1][2][i], k  + noff);
            async_copy_b32(&buf[par ^ 1][3][i], v  + noff);
            async_copy_b32(&buf[par ^ 1][4][i], kk + noff);
            async_copy_b32(&buf[par ^ 1][5][i], bb + noff);
            asm volatile("s_wait_asynccnt 0x6" ::: "memory");
        } else {
            asm volatile("s_wait_asynccnt 0x0" ::: "memory");
        }
        __syncthreads();   // step-t vectors visible to both waves

        const float* rs  = buf[par][0];
        const float* ws  = buf[par][1];
        const float* ks  = buf[par][2];
        const float* vs  = buf[par][3];
        const float* as_ = buf[par][4];
        const float* bs  = buf[par][5];

        float sa = 0.f;
#pragma unroll
        for (int j = 0; j < DDD; ++j) sa += S[j] * as_[j];
        sa = -sa;

        const float vi = vs[i];
        float oi = 0.f;
#pragma unroll
        for (int j = 0; j < DDD; ++j) {
            const float s = S[j] * ws[j] + sa * bs[j] + vi * ks[j];
            S[j] = s;
            oi += s * rs[j];
        }
        o[off] = oi;
        __syncthreads();   // both waves done reading buf[par] before it is refilled
    }
}

// ---------------------------------------------------------------------------
// Post: per head-row GroupNorm + bonus term + gate:
//   og = (GN(o)*gnw + gnb + (sum_d r*k2*r_k) * v) * g
// ---------------------------------------------------------------------------
__global__ __launch_bounds__(64) void rwkv7_post_kernel(
    const float* __restrict__ o, const float* __restrict__ r,
    const float* __restrict__ k2, const float* __restrict__ v,
    const float* __restrict__ r_k, const float* __restrict__ gnw,
    const float* __restrict__ gnb, const float* __restrict__ g,
    float* __restrict__ og)
{
    const int hr = blockIdx.x;
    const int h  = hr % HHH;
    const int m  = hr / HHH;
    const int d  = threadIdx.x;
    const int c  = h * DDD + d;
    const size_t idx = (size_t)m * CCC + c;

    __shared__ float red[DDD];
    const float ov = o[idx];

    red[d] = ov;
    __syncthreads();
    for (int s = 32; s > 0; s >>= 1) { if (d < s) red[d] += red[d + s]; __syncthreads(); }
    const float mu = red[0] * (1.f / DDD);
    __syncthreads();

    const float dv = ov - mu;
    red[d] = dv * dv;
    __syncthreads();
    for (int s = 32; s > 0; s >>= 1) { if (d < s) red[d] += red[d + s]; __syncthreads(); }
    const float var = red[0] * (1.f / DDD);
    __syncthreads();

    const float nrmv = dv * rsqrtf(var + EPSV) * gnw[c] + gnb[c];

    red[d] = r[idx] * k2[idx] * r_k[c];
    __syncthreads();
    for (int s = 32; s > 0; s >>= 1) { if (d < s) red[d] += red[d + s]; __syncthreads(); }
    const float dot = red[0];
    __syncthreads();

    og[idx] = (nrmv + dot * v[idx]) * g[idx];
}

// ---------------------------------------------------------------------------
extern "C" void kernel_launch(void* const* d_in, const int* in_sizes, int n_in,
                              void* d_out, int out_size, void* d_ws, size_t ws_size,
                              hipStream_t stream)
{
    (void)in_sizes; (void)n_in; (void)out_size; (void)ws_size;

    const float* hs     = (const float*)d_in[0];
    const float* x_r    = (const float*)d_in[1];
    const float* x_w    = (const float*)d_in[2];
    const float* x_k    = (const float*)d_in[3];
    const float* x_v    = (const float*)d_in[4];
    const float* x_a    = (const float*)d_in[5];
    /* d_in[6] = x_g: unused by the reference math */
    const float* k_k    = (const float*)d_in[7];
    const float* k_a    = (const float*)d_in[8];
    const float* r_k    = (const float*)d_in[9];
    const float* W_r    = (const float*)d_in[10];
    const float* W_k    = (const float*)d_in[11];
    const float* W_v    = (const float*)d_in[12];
    const float* W_o    = (const float*)d_in[13];
    const float* wla    = (const float*)d_in[14];  // [C,64]
    const float* wlb    = (const float*)d_in[15];  // [64,C]
    const float* wlbias = (const float*)d_in[16];
    const float* ala    = (const float*)d_in[17];  // [C,64]
    const float* alb    = (const float*)d_in[18];  // [64,C]
    const float* albias = (const float*)d_in[19];
    const float* gla    = (const float*)d_in[20];  // [C,128]
    const float* glb    = (const float*)d_in[21];  // [128,C]
    const float* glbias = (const float*)d_in[22];
    const float* gnw    = (const float*)d_in[23];
    const float* gnb    = (const float*)d_in[24];
    float* out          = (float*)d_out;

    float* ws = (float*)d_ws;
    const size_t U = (size_t)MMM * CCC;            // 4M floats per [M,C] buffer
    float* buf_r  = ws + 0 * U;
    float* buf_k  = ws + 1 * U;                    // raw k, becomes k2 in prep
    float* buf_v  = ws + 2 * U;
    float* buf_ew = ws + 3 * U;                    // exp(log_w)
    float* buf_a  = ws + 4 * U;
    float* buf_g  = ws + 5 * U;
    float* buf_kk = ws + 6 * U;
    float* buf_bb = ws + 7 * U;                    // kk * a
    float* buf_o  = ws + 8 * U;
    float* buf_og = ws + 9 * U;
    float* buf_wt = ws + 10 * U;                   // [M,64]
    float* buf_at = buf_wt + (size_t)MMM * 64;     // [M,64]
    float* buf_gt = buf_at + (size_t)MMM * 64;     // [M,128]

    const dim3 blk(256);
    const dim3 gridC(CCC / 64, MMM / 256);         // N=1024 strips
    const dim3 grid64(64 / 64, MMM / 256);         // N=64 strips
    const dim3 grid128(128 / 64, MMM / 256);       // N=128 strips

    // Projections with fused time-shift mixing
    gemm16_wmma<1, 0><<<gridC, blk, 0, stream>>>(hs, x_r, W_r, nullptr, buf_r, MMM, CCC, CCC);
    gemm16_wmma<1, 0><<<gridC, blk, 0, stream>>>(hs, x_k, W_k, nullptr, buf_k, MMM, CCC, CCC);
    gemm16_wmma<1, 0><<<gridC, blk, 0, stream>>>(hs, x_v, W_v, nullptr, buf_v, MMM, CCC, CCC);

    // Decay LoRA: tanh(xw@A) @ B + bias, then exp(-softplus(-z)-0.5)=sigmoid(z)*e^-.5
    gemm16_wmma<1, 1><<<grid64, blk, 0, stream>>>(hs, x_w, wla, nullptr, buf_wt, MMM, CCC, 64);
    gemm16_wmma<0, 3><<<gridC,  blk, 0, stream>>>(buf_wt, nullptr, wlb, wlbias, buf_ew, MMM, 64, CCC);

    // a LoRA: sigmoid((xa@A)@B + bias)
    gemm16_wmma<1, 0><<<grid64, blk, 0, stream>>>(hs, x_a, ala, nullptr, buf_at, MMM, CCC, 64);
    gemm16_wmma<0, 2><<<gridC,  blk, 0, stream>>>(buf_at, nullptr, alb, albias, buf_a, MMM, 64, CCC);

    // g LoRA: sigmoid(r@A) @ B + bias   (activation between the two linears)
    gemm16_wmma<0, 2><<<grid128, blk, 0, stream>>>(buf_r, nullptr, gla, nullptr, buf_gt, MMM, CCC, 128);
    gemm16_wmma<0, 0><<<gridC,   blk, 0, stream>>>(buf_gt, nullptr, glb, glbias, buf_g, MMM, 128, CCC);

    // kk / bb / k2
    rwkv7_prep_kernel<<<MMM * HHH, 64, 0, stream>>>(buf_k, buf_a, k_k, k_a, buf_kk, buf_bb);

    // Sequential scan: one block per (b,h), async double-buffered operands
    rwkv7_scan_kernel<<<BB * HHH, 64, 0, stream>>>(buf_r, buf_ew, buf_k, buf_v,
                                                   buf_kk, buf_bb, buf_o);

    // GroupNorm + bonus + gate
    rwkv7_post_kernel<<<MMM * HHH, 64, 0, stream>>>(buf_o, buf_r, buf_k, buf_v,
                                                    r_k, gnw, gnb, buf_g, buf_og);

    // Output projection
    gemm16_wmma<0, 0><<<gridC, blk, 0, stream>>>(buf_og, nullptr, W_o, nullptr, out, MMM, CCC, CCC);
}